// HeteroDistancePositionEncoding_23888608100655
// MI455X (gfx1250) — compile-verified
//
#include <hip/hip_runtime.h>
#include <hip/hip_bf16.h>

typedef __attribute__((ext_vector_type(2))) float v2f;
typedef __attribute__((ext_vector_type(8))) float v8f;

// ---------------------------------------------------------------------------
// Kernel 1: init reached bitmask (bit b set iff node==b, b<64), zero agg+cnt
// ---------------------------------------------------------------------------
__global__ __launch_bounds__(256) void hdpe_init(unsigned int* __restrict__ reached,
                                                 unsigned int* __restrict__ agg,
                                                 unsigned int* __restrict__ cnt,
                                                 int N) {
    int n = blockIdx.x * blockDim.x + threadIdx.x;
    if (n >= N) return;
    unsigned int lo = (n < 32) ? (1u << n) : 0u;
    unsigned int hi = (n >= 32 && n < 64) ? (1u << (n - 32)) : 0u;
    reached[2 * n + 0] = lo;
    reached[2 * n + 1] = hi;
    agg[2 * n + 0] = 0u;
    agg[2 * n + 1] = 0u;
    cnt[n] = 0u;
}

// ---------------------------------------------------------------------------
// Kernel 2: one BFS propagation level: agg[dst] |= reached[src]
// (stale agg bits from earlier levels are subsets of reached -> harmless)
// ---------------------------------------------------------------------------
__global__ __launch_bounds__(256) void hdpe_propagate(const int* __restrict__ src,
                                                      const int* __restrict__ dst,
                                                      const unsigned int* __restrict__ reached,
                                                      unsigned int* __restrict__ agg,
                                                      int E) {
    int e = blockIdx.x * blockDim.x + threadIdx.x;
    if (e >= E) return;
    int s = src[e];
    unsigned int mx = reached[2 * s + 0];
    unsigned int my = reached[2 * s + 1];
    if (mx | my) {
        int d = dst[e];
        if (mx) atomicOr(&agg[2 * d + 0], mx);
        if (my) atomicOr(&agg[2 * d + 1], my);
    }
}

// ---------------------------------------------------------------------------
// Kernel 3: fold level k: count newly reached bits, merge into reached.
// cnt packs level-1..3 counts into bytes 0..2 (each count <= 64 < 256).
// ---------------------------------------------------------------------------
__global__ __launch_bounds__(256) void hdpe_update(unsigned int* __restrict__ reached,
                                                   const unsigned int* __restrict__ agg,
                                                   unsigned int* __restrict__ cnt,
                                                   int N, int k) {
    int n = blockIdx.x * blockDim.x + threadIdx.x;
    if (n >= N) return;
    unsigned int rx = reached[2 * n + 0], ry = reached[2 * n + 1];
    unsigned int ax = agg[2 * n + 0],     ay = agg[2 * n + 1];
    unsigned int nwx = ax & ~rx, nwy = ay & ~ry;
    unsigned int c = (unsigned int)(__popc(nwx) + __popc(nwy));
    if (c) cnt[n] += c << (8 * (k - 1));
    if (nwx) reached[2 * n + 0] = rx | nwx;
    if (nwy) reached[2 * n + 1] = ry | nwy;
}

// ---------------------------------------------------------------------------
// Kernel 4: copy x [N,128] into out[:, 0:128] (out row stride 160), float4
// ---------------------------------------------------------------------------
__global__ __launch_bounds__(256) void hdpe_copy_x(const float4* __restrict__ x4,
                                                   float* __restrict__ out,
                                                   long long total4) {
    long long i = (long long)blockIdx.x * blockDim.x + threadIdx.x;
    if (i >= total4) return;
    long long n = i >> 5;        // node
    int j = (int)(i & 31);       // float4 index within row (128/4)
    float4 v = x4[i];
    *(float4*)(out + n * 160 + j * 4) = v;
}

// ---------------------------------------------------------------------------
// Kernel 5: pos_enc via V_WMMA_F32_16X16X4_F32.
//   A (16x4): per node row m: [ (node<64), cnt_l1, cnt_l2, cnt_l3 ]
//   B (4x16): embAdj[d][col] = emb[d][col] - emb[4][col]
//   C (16x16): 64 * emb[4][col]  (bias, all rows identical)
// One wave -> 16 nodes x 32 pe columns (two col-half WMMAs).
// A layout: lanes 0-15 hold (K0,K1), lanes 16-31 hold (K2,K3), M = lane%16.
// B layout: vgpr0 = rows K=0 (lanes 0-15) / K=2 (lanes 16-31); vgpr1 = K=1/K=3.
// D layout: vgpr v, lane l -> M = v + (l<16?0:8), Ncol = l%16.
// Epilogue: wave-uniform fast path (full tile) stores 8 rows off one base
// address with immediate offsets; guarded path only for the last tile.
// ---------------------------------------------------------------------------
__global__ __launch_bounds__(256) void hdpe_pe_wmma(const unsigned int* __restrict__ cnt,
                                                    const float* __restrict__ emb,
                                                    float* __restrict__ out,
                                                    int N, int ntiles) {
    int gtid = blockIdx.x * blockDim.x + threadIdx.x;
    int wave = gtid >> 5;
    if (wave >= ntiles) return;              // wave-uniform: EXEC stays all-1s
    int lane = threadIdx.x & 31;
    int half = lane >> 4;                    // 0: K=0/1, 1: K=2/3
    int l = lane & 15;

    int tile = wave * 16;
    int node = tile + l;
    int cn = node < N ? node : (N - 1);      // clamp loads; stores guarded later

    unsigned int c = cnt[cn];
    float c1 = (float)(c & 0xffu);
    float c2 = (float)((c >> 8) & 0xffu);
    float c3 = (float)((c >> 16) & 0xffu);
    float ind0 = (node < 64) ? 1.0f : 0.0f;

    v2f a;
    if (half == 0) { a.x = ind0; a.y = c1; }
    else           { a.x = c2;   a.y = c3; }

    int k0 = half ? 2 : 0;
    int k1 = half ? 3 : 1;

    bool full_tile = (tile + 16 <= N);       // wave-uniform

    #pragma unroll
    for (int colbase = 0; colbase < 32; colbase += 16) {
        int col = colbase + l;
        float e4 = emb[4 * 32 + col];
        v2f b;
        b.x = emb[k0 * 32 + col] - e4;
        b.y = emb[k1 * 32 + col] - e4;

        v8f cacc;
        #pragma unroll
        for (int v = 0; v < 8; ++v) cacc[v] = 64.0f * e4;

        v8f d = __builtin_amdgcn_wmma_f32_16x16x4_f32(
            /*neg_a=*/false, a, /*neg_b=*/false, b,
            /*c_mod=*/(short)0, cacc, /*reuse_a=*/false, /*reuse_b=*/false);

        // Base of this lane's first output element: row (tile + half*8), col.
        float* rowp = out + (long long)(tile + half * 8) * 160 + 128 + col;
        if (full_tile) {
            #pragma unroll
            for (int v = 0; v < 8; ++v) rowp[v * 160] = d[v];   // imm offs v*640B
        } else {
            #pragma unroll
            for (int v = 0; v < 8; ++v) {
                if (tile + half * 8 + v < N) rowp[v * 160] = d[v];
            }
        }
    }
}

// ---------------------------------------------------------------------------
extern "C" void kernel_launch(void* const* d_in, const int* in_sizes, int n_in,
                              void* d_out, int out_size, void* d_ws, size_t ws_size,
                              hipStream_t stream) {
    const float* x   = (const float*)d_in[0];   // [N,128]
    const float* emb = (const float*)d_in[1];   // [5,32]
    const int*   src = (const int*)d_in[2];     // [E]
    const int*   dst = (const int*)d_in[3];     // [E]
    float* out = (float*)d_out;                 // [N,160]

    int N = in_sizes[0] / 128;
    int E = in_sizes[2];

    unsigned int* reached = (unsigned int*)d_ws;          // 2N u32
    unsigned int* agg     = reached + 2 * (size_t)N;      // 2N u32
    unsigned int* cnt     = agg + 2 * (size_t)N;          // N  u32

    const int T = 256;
    int gN = (N + T - 1) / T;
    int gE = (E + T - 1) / T;

    hdpe_init<<<gN, T, 0, stream>>>(reached, agg, cnt, N);

    // Only 3 levels needed: level-4-reached and unreached share emb[4].
    for (int k = 1; k <= 3; ++k) {
        hdpe_propagate<<<gE, T, 0, stream>>>(src, dst, reached, agg, E);
        hdpe_update<<<gN, T, 0, stream>>>(reached, agg, cnt, N, k);
    }

    long long total4 = (long long)N * 32;     // float4 chunks of x
    int gX = (int)((total4 + T - 1) / T);
    hdpe_copy_x<<<gX, T, 0, stream>>>((const float4*)x, out, total4);

    int ntiles = (N + 15) / 16;               // 16 nodes per wave
    int waves_per_block = T / 32;
    int gW = (ntiles + waves_per_block - 1) / waves_per_block;
    hdpe_pe_wmma<<<gW, T, 0, stream>>>(cnt, emb, out, N, ntiles);
}